// ChamferDistance_17849884082443
// MI455X (gfx1250) — compile-verified
//
#include <hip/hip_runtime.h>

typedef float v2f __attribute__((ext_vector_type(2)));
typedef float v8f __attribute__((ext_vector_type(8)));

#define BATCH          4
#define NPTS           8192
#define CHUNKS         8
#define WAVES_PER_BLK  8

// Initialize output to +inf bit pattern so uint atomicMin works
// (all chamfer distances are >= 0, so uint order == float order).
__global__ __launch_bounds__(256) void chamfer_init(unsigned int* __restrict__ out, int n) {
    int i = blockIdx.x * blockDim.x + threadIdx.x;
    if (i < n) out[i] = 0x7F800000u;  // +inf
}

// One wave owns a 32-row strip of `rows` (two 16-row WMMA tiles) and scans a
// chunk of `cols`. Per 16-column tile we issue TWO V_WMMA_F32_16X16X4_F32 that
// share the same B operand (and the per-lane ||q||^2), amortizing the column
// load, n2 computation and operand selects over 512 output elements:
//   A row  = (-2x, -2y, -2z, ||p||^2)   (K = 4)
//   B col  = ( qx,  qy,  qz, 1 )
// so C[n][m] = ||p||^2 - 2 p.q ; then d = C + ||q||^2, clamp+min via v_med3.
__global__ __launch_bounds__(256) void chamfer_wmma(const float* __restrict__ rows,
                                                    const float* __restrict__ cols,
                                                    unsigned int* __restrict__ outmin,
                                                    int NR, int NC) {
    const int lane  = threadIdx.x & 31;
    const int wave  = blockIdx.x * WAVES_PER_BLK + (threadIdx.x >> 5);
    const int stripsPerBatch = NR >> 5;                 // 32 rows per wave
    const int chunk = wave % CHUNKS;
    const int strip = (wave / CHUNKS) % stripsPerBatch;
    const int batch = wave / (CHUNKS * stripsPerBatch);
    const int rowbase = strip << 5;
    const bool hi = (lane >= 16);

    // ---- A tiles: 2 x (16 rows x K=4). ISA layout: lanes 0-15 hold K0,K1 in
    // the two VGPRs; lanes 16-31 hold K2,K3. Row M = lane & 15.
    v2f av0, av1;
    {
        const int pr = rowbase + (lane & 15);
        const float* rp = rows + 3 * (size_t)(batch * NR + pr);
        float x0 = rp[0], x1 = rp[1], x2 = rp[2];
        float n1 = fmaf(x0, x0, fmaf(x1, x1, x2 * x2));
        av0.x = hi ? (-2.0f * x2) : (-2.0f * x0);
        av0.y = hi ? n1           : (-2.0f * x1);
        const float* rq = rp + 3 * 16;                  // second 16-row tile
        x0 = rq[0]; x1 = rq[1]; x2 = rq[2];
        n1 = fmaf(x0, x0, fmaf(x1, x1, x2 * x2));
        av1.x = hi ? (-2.0f * x2) : (-2.0f * x0);
        av1.y = hi ? n1           : (-2.0f * x1);
    }

    float rm0[8], rm1[8];
#pragma unroll
    for (int r = 0; r < 8; ++r) { rm0[r] = __builtin_inff(); rm1[r] = __builtin_inff(); }

    const int colsPerChunk = NC / CHUNKS;
    const int c0     = chunk * colsPerChunk;
    const int ntiles = colsPerChunk >> 4;
    const float* cbase = cols + 3 * (size_t)batch * NC;

#pragma unroll 2
    for (int t = 0; t < ntiles; ++t) {
        const int col = c0 + (t << 4) + (lane & 15);
        const float* cp = cbase + 3 * (size_t)col;
        const float y0 = cp[0], y1 = cp[1], y2 = cp[2];
        const float n2 = fmaf(y0, y0, fmaf(y1, y1, y2 * y2));

        // B tile 4x16: lanes 0-15 hold rows K0,K1; lanes 16-31 hold rows K2,K3.
        v2f bv;
        bv.x = hi ? y2   : y0;
        bv.y = hi ? 1.0f : y1;

        v8f c0v = {}, c1v = {};
        c0v = __builtin_amdgcn_wmma_f32_16x16x4_f32(
                false, av0, false, bv, (short)0, c0v, false, false);
        c1v = __builtin_amdgcn_wmma_f32_16x16x4_f32(
                false, av1, false, bv, (short)0, c1v, false, false);

#pragma unroll
        for (int r = 0; r < 8; ++r) {
            const float d0 = c0v[r] + n2;                      // full sq. distance
            rm0[r] = __builtin_amdgcn_fmed3f(0.0f, d0, rm0[r]); // = min(max(d,0), rm)
            const float d1 = c1v[r] + n2;
            rm1[r] = __builtin_amdgcn_fmed3f(0.0f, d1, rm1[r]);
        }
    }

    // Per-row min: reduce across the 16 lanes of each half-wave.
    // Accumulator layout: vgpr r -> row r (lanes 0-15), row r+8 (lanes 16-31).
    unsigned int* op = outmin + (size_t)batch * NR + rowbase;
#pragma unroll
    for (int r = 0; r < 8; ++r) {
        float v = rm0[r];
        v = fminf(v, __shfl_xor(v, 1, 32));
        v = fminf(v, __shfl_xor(v, 2, 32));
        v = fminf(v, __shfl_xor(v, 4, 32));
        v = fminf(v, __shfl_xor(v, 8, 32));
        if (lane == 0)  atomicMin(op + r,      __float_as_uint(v));
        if (lane == 16) atomicMin(op + r + 8,  __float_as_uint(v));

        float w = rm1[r];
        w = fminf(w, __shfl_xor(w, 1, 32));
        w = fminf(w, __shfl_xor(w, 2, 32));
        w = fminf(w, __shfl_xor(w, 4, 32));
        w = fminf(w, __shfl_xor(w, 8, 32));
        if (lane == 0)  atomicMin(op + 16 + r,     __float_as_uint(w));
        if (lane == 16) atomicMin(op + 16 + r + 8, __float_as_uint(w));
    }
}

extern "C" void kernel_launch(void* const* d_in, const int* in_sizes, int n_in,
                              void* d_out, int out_size, void* d_ws, size_t ws_size,
                              hipStream_t stream) {
    const float* xyz1 = (const float*)d_in[0];   // (B, N, 3) fp32
    const float* xyz2 = (const float*)d_in[1];   // (B, M, 3) fp32
    unsigned int* uout = (unsigned int*)d_out;   // dist1 (B*N) then dist2 (B*M)

    const int total = out_size;                  // 2 * B * NPTS
    chamfer_init<<<(total + 255) / 256, 256, 0, stream>>>(uout, total);

    const int stripsPerBatch = NPTS / 32;
    const int wavesPerPass   = BATCH * stripsPerBatch * CHUNKS;   // 8192
    const int blocks         = wavesPerPass / WAVES_PER_BLK;      // 1024

    // dist1: rows = xyz1, min over xyz2
    chamfer_wmma<<<blocks, 256, 0, stream>>>(xyz1, xyz2, uout, NPTS, NPTS);
    // dist2: rows = xyz2, min over xyz1
    chamfer_wmma<<<blocks, 256, 0, stream>>>(xyz2, xyz1, uout + BATCH * NPTS, NPTS, NPTS);
}